// TripletCenterLoss_27204322853527
// MI455X (gfx1250) — compile-verified
//
#include <hip/hip_runtime.h>
#include <math.h>

typedef float v2f __attribute__((ext_vector_type(2)));
typedef float v8f __attribute__((ext_vector_type(8)));

#define B_SZ   1024
#define D_SZ   256
#define C_SZ   55
#define CPAD   64
#define MARGIN 0.2f

// ---------------------------------------------------------------------------
// Kernel 1: class-present flags. One block of 1024 threads.
// present[c] = 1 iff class c appears in targets. Benign race (all write 1).
// ---------------------------------------------------------------------------
__global__ void k_flags(const int* __restrict__ targets, int* __restrict__ present) {
    int i = threadIdx.x;
    if (i < CPAD) present[i] = 0;
    __syncthreads();
    int t = targets[i];
    present[t] = 1;
}

// ---------------------------------------------------------------------------
// Kernel 2: squared norms. One wave (32 lanes) per row, wave32 shuffle reduce.
// Rows 0..1023 -> x2 (inputs), rows 1024..1087 -> c2 (centers, zero-padded).
// ---------------------------------------------------------------------------
__global__ void k_norms(const float* __restrict__ inputs,
                        const float* __restrict__ centers,
                        float* __restrict__ x2, float* __restrict__ c2) {
    int wave = (int)((blockIdx.x * blockDim.x + threadIdx.x) >> 5);
    int lane = threadIdx.x & 31;
    if (wave < B_SZ) {
        const float* row = inputs + (size_t)wave * D_SZ;
        float s = 0.f;
        #pragma unroll
        for (int k = lane; k < D_SZ; k += 32) { float v = row[k]; s += v * v; }
        #pragma unroll
        for (int off = 16; off; off >>= 1) s += __shfl_xor(s, off, 32);
        if (lane == 0) x2[wave] = s;
    } else if (wave < B_SZ + CPAD) {
        int c = wave - B_SZ;
        float s = 0.f;
        if (c < C_SZ) {
            const float* row = centers + (size_t)c * D_SZ;
            #pragma unroll
            for (int k = lane; k < D_SZ; k += 32) { float v = row[k]; s += v * v; }
            #pragma unroll
            for (int off = 16; off; off >>= 1) s += __shfl_xor(s, off, 32);
        }
        if (lane == 0) c2[c] = s;   // zero for pad classes 55..63
    }
}

// ---------------------------------------------------------------------------
// Kernel 3: dsmall[i,c] = sqrt(clamp(x2[i] - 2*(inputs@centers^T)[i,c] + c2[c]))
// via V_WMMA_F32_16X16X4_F32. One wave per 16x16 output tile.
// 64 row-tiles x 4 col-tiles = 256 waves; block = 256 threads (8 waves).
//
// fp32 16x16x4 fragment layout (ISA 7.12.2):
//   A (16x4):  lane&15 = M, VGPR{0,1} = K{0,1} (+2 for lanes 16..31)
//   B (4x16):  lane&15 = N, VGPR{0,1} = K{0,1} (+2 for lanes 16..31)
//   C/D 16x16: VGPR v -> M = v + 8*(lane>=16), N = lane&15
// ---------------------------------------------------------------------------
__global__ void k_dist_wmma(const float* __restrict__ inputs,
                            const float* __restrict__ centers,
                            const float* __restrict__ x2,
                            const float* __restrict__ c2,
                            float* __restrict__ dist) {
    const int wave  = (int)((blockIdx.x * blockDim.x + threadIdx.x) >> 5);
    const int lane  = threadIdx.x & 31;
    const int half  = lane >> 4;       // 0: K=0,1   1: K=2,3
    const int l15   = lane & 15;
    const int tileM = wave >> 2;       // 0..63
    const int tileN = wave & 3;        // 0..3

    const int arow = tileM * 16 + l15;                 // A: M index
    const int bcol = tileN * 16 + l15;                 // B: N index (class)
    const int brow = bcol < C_SZ ? bcol : (C_SZ - 1);  // pad cols reuse row 54
                                                       // (never read; avoids
                                                       //  divergence near WMMA)
    const float* aptr = inputs  + (size_t)arow * D_SZ + half * 2;
    const float* bptr = centers + (size_t)brow * D_SZ + half * 2;

    v8f acc = {};
    #pragma unroll 8
    for (int kb = 0; kb < D_SZ; kb += 4) {
        v2f a = *(const v2f*)(aptr + kb);
        v2f b = *(const v2f*)(bptr + kb);
        // D = A(16x4) * B(4x16) + C, full fp32
        acc = __builtin_amdgcn_wmma_f32_16x16x4_f32(
            /*neg_a=*/false, a, /*neg_b=*/false, b,
            /*c_mod=*/(short)0, acc, /*reuse_a=*/false, /*reuse_b=*/false);
    }

    const float cn = c2[bcol];
    #pragma unroll
    for (int v = 0; v < 8; ++v) {
        int m   = tileM * 16 + v + half * 8;
        float g = acc[v];
        float d2 = x2[m] - 2.0f * g + cn;
        d2 = d2 > 1e-12f ? d2 : 1e-12f;
        dist[(size_t)m * CPAD + bcol] = sqrtf(d2);
    }
}

// ---------------------------------------------------------------------------
// Kernel 4: hard-positive / hard-negative mining + deterministic reduction.
// One block of 1024 threads; fixed-order LDS tree (no float atomics).
// ---------------------------------------------------------------------------
__global__ void k_reduce(const float* __restrict__ dist,
                         const int* __restrict__ targets,
                         const int* __restrict__ present,
                         float* __restrict__ out) {
    __shared__ float s_loss[B_SZ];
    __shared__ float s_diff[B_SZ];

    int i = threadIdx.x;
    int t = targets[i];
    const float* row = dist + (size_t)i * CPAD;

    float ap = row[t];              // all same-class columns equal this value
    float an = INFINITY;
    for (int c = 0; c < C_SZ; ++c) {
        if (c != t && present[c]) {
            float d = row[c];
            an = d < an ? d : an;
        }
    }

    float l = ap - an + MARGIN;
    s_loss[i] = l > 0.f ? l : 0.f;  // relu
    s_diff[i] = an - ap;
    __syncthreads();

    for (int off = B_SZ / 2; off; off >>= 1) {
        if (i < off) {
            s_loss[i] += s_loss[i + off];
            s_diff[i] += s_diff[i + off];
        }
        __syncthreads();
    }
    if (i == 0) {
        out[0] = s_loss[0] * (1.0f / (float)B_SZ);  // loss = mean(relu(...))
        out[1] = s_diff[0] * (1.0f / (float)B_SZ);  // prec = sum(an-ap)/B
    }
}

// ---------------------------------------------------------------------------
// Host launcher
// ---------------------------------------------------------------------------
extern "C" void kernel_launch(void* const* d_in, const int* in_sizes, int n_in,
                              void* d_out, int out_size, void* d_ws, size_t ws_size,
                              hipStream_t stream) {
    const float* inputs  = (const float*)d_in[0];   // [1024, 256] f32
    const int*   targets = (const int*)  d_in[1];   // [1024] int
    const float* centers = (const float*)d_in[2];   // [55, 256] f32
    float* out = (float*)d_out;                     // [loss, prec]

    // workspace layout
    float* dist    = (float*)d_ws;                  // 1024*64 f32 (256 KB)
    float* x2      = dist + (size_t)B_SZ * CPAD;    // 1024 f32
    float* c2      = x2 + B_SZ;                     // 64 f32 (zero-padded)
    int*   present = (int*)(c2 + CPAD);             // 64 int

    k_flags<<<1, B_SZ, 0, stream>>>(targets, present);
    k_norms<<<(B_SZ + CPAD) / 8, 256, 0, stream>>>(inputs, centers, x2, c2);
    k_dist_wmma<<<32, 256, 0, stream>>>(inputs, centers, x2, c2, dist);
    k_reduce<<<1, B_SZ, 0, stream>>>(dist, targets, present, out);
}